// Model_1580547965011
// MI455X (gfx1250) — compile-verified
//
#include <hip/hip_runtime.h>
#include <math.h>

typedef __attribute__((ext_vector_type(2))) float v2f;
typedef __attribute__((ext_vector_type(8))) float v8f;

#define VOCAB 128000
#define EMB   32
#define HALF  16
#define BATCH 1024
#define SLEN  50

// ---------------------------------------------------------------------------
// Kernel A: hidden = [attn-pooled ; last-token] @ out_w.T + out_b   (1024 x 32)
// One wave32 per batch row; lane j == channel j (EMB == warpSize == 32).
// ---------------------------------------------------------------------------
__global__ __launch_bounds__(128) void hidden_kernel(
    const int*   __restrict__ text,  const float* __restrict__ emb,
    const float* __restrict__ fc0_w, const float* __restrict__ fc0_b,
    const float* __restrict__ att_wq,const float* __restrict__ att_wk,
    const float* __restrict__ att_v, const float* __restrict__ out_w,
    const float* __restrict__ out_b, float*       __restrict__ hidden)
{
    __shared__ float e_lds[4][SLEN][EMB];   // embedded rows, per wave
    __shared__ float sc_lds[4][SLEN];       // masked scores
    __shared__ float sg_lds[4][EMB];        // pooled vector

    const int lane = threadIdx.x & 31;
    const int wave = threadIdx.x >> 5;
    const int b    = blockIdx.x * 4 + wave;

    float w0[HALF];
    #pragma unroll
    for (int h = 0; h < HALF; ++h) w0[h] = fc0_w[lane * HALF + h];
    const float b0 = fc0_b[lane];

    float wqk[EMB];                          // (Wq + Wk) row for this channel
    #pragma unroll
    for (int k = 0; k < EMB; ++k) wqk[k] = att_wq[lane*EMB+k] + att_wk[lane*EMB+k];
    const float vj = att_v[lane];

    unsigned long long validmask = 0ull;
    for (int s = 0; s < SLEN; ++s) {
        const int t = text[b * SLEN + s];
        const bool valid = (t != 0) && (t != 1);
        if (valid) validmask |= (1ull << s);
        const float* er = emb + (long long)t * HALF;
        float acc = 0.f;
        #pragma unroll
        for (int h = 0; h < HALF; ++h) acc += er[h] * w0[h];
        e_lds[wave][s][lane] = valid ? (acc + b0) : 0.f;
    }
    __syncthreads();

    // scores[s] = sum_j tanh(sum_k e[s][k]*wqk[j][k]) * v[j]  (wave reduction)
    for (int s = 0; s < SLEN; ++s) {
        float z = 0.f;
        #pragma unroll
        for (int k = 0; k < EMB; ++k) z += e_lds[wave][s][k] * wqk[k];
        float p = tanhf(z) * vj;
        #pragma unroll
        for (int off = 16; off > 0; off >>= 1) p += __shfl_xor(p, off, 32);
        if (lane == 0)
            sc_lds[wave][s] = ((validmask >> s) & 1ull) ? p : -1.0e9f;
    }
    __syncthreads();

    // masked softmax + pooled sum
    float m = -1.0e30f;
    for (int s = 0; s < SLEN; ++s) m = fmaxf(m, sc_lds[wave][s]);
    float denom = 0.f;
    for (int s = 0; s < SLEN; ++s) denom += __expf(sc_lds[wave][s] - m);
    const float inv = 1.f / denom;
    float sg = 0.f;
    for (int s = 0; s < SLEN; ++s)
        sg += __expf(sc_lds[wave][s] - m) * inv * e_lds[wave][s][lane];
    sg_lds[wave][lane] = sg;
    __syncthreads();

    float h = out_b[lane];
    #pragma unroll
    for (int k = 0; k < EMB; ++k) h += sg_lds[wave][k]           * out_w[lane*2*EMB + k];
    #pragma unroll
    for (int k = 0; k < EMB; ++k) h += e_lds[wave][SLEN-1][k]    * out_w[lane*2*EMB + EMB + k];
    hidden[b * EMB + lane] = h;
}

// ---------------------------------------------------------------------------
// Kernel B: out[b][v] = hidden[b] . item_vecs[v],
//           item_vecs = emb_weight @ fc0_w.T + fc0_b computed per N-tile.
// 16x16 output tiles via 8 chained V_WMMA_F32_16X16X4_F32 (K = 32).
//   A 16x4 layout: lane L -> M = L%16; {a.x,a.y} = A[M][kb], A[M][kb+1], kb=2*(L/16)
//   B  4x16 layout (symmetric): lane L -> N = L%16; {b.x,b.y} = B[kb][N], B[kb+1][N]
//   C/D 16x16:     lane L -> N = L%16; VGPR r -> M = r + 8*(L/16)
// Output stores are non-temporal: 524 MB pure streaming write that is never
// re-read; keep L2 for the 8000x re-read `hidden` (128 KB) and emb tiles.
// ---------------------------------------------------------------------------
__global__ __launch_bounds__(256) void out_gemm(
    const float* __restrict__ hidden, const float* __restrict__ emb,
    const float* __restrict__ fc0_w,  const float* __restrict__ fc0_b,
    float*       __restrict__ out)
{
    __shared__ float Bt[EMB][16];            // Bt[k][n] = item_vecs[n0+n][k]
    const int tid = threadIdx.x;
    const int n0  = blockIdx.x * 16;

    // Build the B-tile (16 vocab rows x 32 channels) once per block.
    for (int e = tid; e < EMB * 16; e += 256) {
        const int k = e >> 4, n = e & 15;
        const float* er = emb + (long long)(n0 + n) * HALF;
        const float* wr = fc0_w + k * HALF;
        float acc = fc0_b[k];
        #pragma unroll
        for (int h = 0; h < HALF; ++h) acc += er[h] * wr[h];
        Bt[k][n] = acc;
    }
    __syncthreads();

    const int lane = tid & 31;
    const int wave = tid >> 5;
    const int hlf  = lane >> 4;
    const int kb   = 2 * hlf;
    const int n    = lane & 15;

    // B fragments are loop-invariant across M-tiles: hoist into registers.
    v2f bf[8];
    #pragma unroll
    for (int st = 0; st < 8; ++st) {
        bf[st].x = Bt[4*st + kb    ][n];
        bf[st].y = Bt[4*st + kb + 1][n];
    }

    for (int mt = wave; mt < BATCH / 16; mt += 8) {
        const int row = mt * 16 + n;         // A-matrix M index == lane%16
        v2f af[8];
        #pragma unroll
        for (int st = 0; st < 8; ++st)
            af[st] = *(const v2f*)(hidden + row * EMB + 4*st + kb);

        v8f c = {};
        #pragma unroll
        for (int st = 0; st < 8; ++st)
            c = __builtin_amdgcn_wmma_f32_16x16x4_f32(
                    false, af[st], false, bf[st], (short)0, c, false, false);

        #pragma unroll
        for (int r = 0; r < 8; ++r) {
            const int grow = mt * 16 + r + 8 * hlf;
            __builtin_nontemporal_store(c[r], &out[(size_t)grow * VOCAB + n0 + n]);
        }
    }
}

extern "C" void kernel_launch(void* const* d_in, const int* in_sizes, int n_in,
                              void* d_out, int out_size, void* d_ws, size_t ws_size,
                              hipStream_t stream) {
    const int*   text  = (const int*)  d_in[0];
    const float* emb   = (const float*)d_in[1];
    const float* fc0_w = (const float*)d_in[2];
    const float* fc0_b = (const float*)d_in[3];
    const float* wq    = (const float*)d_in[4];
    const float* wk    = (const float*)d_in[5];
    const float* av    = (const float*)d_in[6];
    const float* ow    = (const float*)d_in[7];
    const float* ob    = (const float*)d_in[8];
    float* hidden = (float*)d_ws;            // 1024*32*4 = 128 KB scratch
    float* outp   = (float*)d_out;

    hidden_kernel<<<BATCH / 4, 128, 0, stream>>>(text, emb, fc0_w, fc0_b,
                                                 wq, wk, av, ow, ob, hidden);
    out_gemm<<<VOCAB / 16, 256, 0, stream>>>(hidden, emb, fc0_w, fc0_b, outp);
}